// BipartiteGraphMatcher_5446018531384
// MI455X (gfx1250) — compile-verified
//
#include <hip/hip_runtime.h>
#include <hip/hip_bf16.h>

typedef __attribute__((ext_vector_type(2))) float v2f;
typedef __attribute__((ext_vector_type(8))) float v8f;

#define SINK_ITERS 10000

// Sum of 129 floats in LDS, computed redundantly by every wave (all lanes get result).
__device__ __forceinline__ float wave_reduce_sum129(const float* v, int lane) {
    float s = v[lane] + v[lane + 32] + v[lane + 64] + v[lane + 96];
    if (lane == 0) s += v[128];
#pragma unroll
    for (int o = 16; o > 0; o >>= 1) s += __shfl_xor(s, o, 32);
    return s;
}

// One workgroup per batch. 256 threads = 8 waves; wave w owns row-block and
// col-block [16w, 16w+16) of the 128x128 core. K = exp(cost) lives entirely in
// VGPRs as WMMA B-fragments; the scaling vectors a=exp(u), b=exp(v) live in LDS.
__global__ __launch_bounds__(256, 1)
void sinkhorn_wmma_kernel(const float* __restrict__ cost,
                          const float* __restrict__ bin_score,
                          float* __restrict__ out)
{
    const int batch  = blockIdx.x;
    const int tid    = threadIdx.x;
    const int lane   = tid & 31;
    const int wave   = tid >> 5;
    const int nl     = lane & 15;         // N index within 16-wide tile
    const int h      = (lane >> 4) << 1;  // K-pair select: 0 (lanes 0-15), 2 (lanes 16-31)
    const int base16 = wave * 16;

    const float alpha = bin_score[0];
    const float ea    = __expf(alpha);    // all dustbin K entries equal exp(alpha)

    const float MU_CORE = 1.0f / 256.0f;  // exp(norm), m=n=128
    const float MU_BIN  = 0.5f;           // exp(log(n)+norm) = 128/256
    const float NU_CORE = 1.0f / 256.0f;
    const float NU_BIN  = 0.5f;
    const float NORM    = -5.545177444479562f; // -log(256)

    __shared__ __align__(16) float lds_a[132];  // a = exp(u), 129 entries
    __shared__ __align__(16) float lds_b[132];  // b = exp(v), 129 entries

    const float* Kc = cost + batch * (128 * 128);

    // ---- Preload K = exp(cost) as register-resident WMMA B-fragments ----
    // frRow[s]: B[k][n] = K(base16+n, k)   -> D[m,n] = sum_k b[k]*K(row,k) = y[row]
    // frCol[s]: B[k][n] = K(k, base16+n)   -> D[m,n] = sum_k a[k]*K(k,col) = c[col]
    v2f frRow[32];
    v2f frCol[32];
#pragma unroll
    for (int s = 0; s < 32; ++s) {
        const int k = 4 * s + h;
        const float* rp = Kc + (base16 + nl) * 128 + k;
        frRow[s].x = __expf(rp[0]);
        frRow[s].y = __expf(rp[1]);
        const float* cp = Kc + k * 128 + (base16 + nl);
        frCol[s].x = __expf(cp[0]);
        frCol[s].y = __expf(cp[128]);
    }

    // init: v = 0  =>  b = 1
    for (int i = tid; i < 129; i += 256) lds_b[i] = 1.0f;

#pragma unroll 1
    for (int it = 0; it < SINK_ITERS; ++it) {
        // ---------- row update: a = mu / (K b) ----------
        __syncthreads();                               // lds_b complete
        const float sumb = wave_reduce_sum129(lds_b, lane);
        const float b128 = lds_b[128];

        v8f acc = {0.f, 0.f, 0.f, 0.f, 0.f, 0.f, 0.f, 0.f};
#pragma unroll
        for (int s = 0; s < 32; ++s) {
            // A operand: b-segment broadcast over all 16 rows (lane-uniform per half)
            v2f va = *(const v2f*)&lds_b[4 * s + h];
            acc = __builtin_amdgcn_wmma_f32_16x16x4_f32(
                false, va, false, frRow[s], (short)0, acc, false, false);
        }
        const float y  = acc[0] + ea * b128;           // + dustbin column term
        const float av = MU_CORE * __builtin_amdgcn_rcpf(y);
        if (lane < 16) lds_a[base16 + lane] = av;
        if (tid == 0)  lds_a[128] = MU_BIN * __builtin_amdgcn_rcpf(ea * sumb);

        // ---------- col update: b = nu / (K^T a) ----------
        __syncthreads();                               // lds_a complete, lds_b reads done
        const float suma = wave_reduce_sum129(lds_a, lane);
        const float a128 = lds_a[128];

        v8f acc2 = {0.f, 0.f, 0.f, 0.f, 0.f, 0.f, 0.f, 0.f};
#pragma unroll
        for (int s = 0; s < 32; ++s) {
            v2f va = *(const v2f*)&lds_a[4 * s + h];
            acc2 = __builtin_amdgcn_wmma_f32_16x16x4_f32(
                false, va, false, frCol[s], (short)0, acc2, false, false);
        }
        const float c  = acc2[0] + ea * a128;          // + dustbin row term
        const float bv = NU_CORE * __builtin_amdgcn_rcpf(c);
        if (lane < 16) lds_b[base16 + lane] = bv;
        if (tid == 0)  lds_b[128] = NU_BIN * __builtin_amdgcn_rcpf(ea * suma);
    }

    // ---------- epilogue: Z = Z0 + log(a) + log(b) - norm ----------
    __syncthreads();
    if (tid < 129) {
        lds_a[tid] = __logf(lds_a[tid]);   // u
        lds_b[tid] = __logf(lds_b[tid]);   // v
    }
    __syncthreads();

    float* ob = out + batch * (129 * 129);
    for (int idx = tid; idx < 129 * 129; idx += 256) {
        const int i = idx / 129;
        const int j = idx - i * 129;
        const float z0 = (i < 128 && j < 128) ? Kc[i * 128 + j] : alpha;
        ob[idx] = z0 + lds_a[i] + lds_b[j] - NORM;
    }
}

extern "C" void kernel_launch(void* const* d_in, const int* in_sizes, int n_in,
                              void* d_out, int out_size, void* d_ws, size_t ws_size,
                              hipStream_t stream) {
    (void)in_sizes; (void)n_in; (void)out_size; (void)d_ws; (void)ws_size;
    const float* cost = (const float*)d_in[0];   // (4,128,128) f32
    const float* bin  = (const float*)d_in[1];   // scalar f32
    float* out = (float*)d_out;                  // (4,129,129) f32
    sinkhorn_wmma_kernel<<<dim3(4), dim3(256), 0, stream>>>(cost, bin, out);
}